// PetriNetAlignmentPredictor_86741159510255
// MI455X (gfx1250) — compile-verified
//
#include <hip/hip_runtime.h>
#include <math.h>

#define B_SZ   4096
#define NUM_M  128
#define NUM_T  64
#define HID    64
#define LDPAD  132   // 128 + 4 : bank stride 4 -> conflict-free fragment loads
#define GRP    4     // batch elements per workgroup (amortizes Omega streaming)

typedef __attribute__((ext_vector_type(2))) float v2f;
typedef __attribute__((ext_vector_type(8))) float v8f;

// ---------------------------------------------------------------------------
// Kernel 1: alphas = sigmoid(relu([vc,vt] @ W1 + b1) @ W2 + b2)   [B, 64]
// ---------------------------------------------------------------------------
__global__ __launch_bounds__(64) void alphas_kernel(
    const float* __restrict__ vc, const float* __restrict__ vt,
    const float* __restrict__ W1, const float* __restrict__ b1,
    const float* __restrict__ W2, const float* __restrict__ b2,
    float* __restrict__ alphas_out)
{
    __shared__ float xs[2 * NUM_M];
    __shared__ float hs[HID];
    const int b = blockIdx.x;
    const int t = threadIdx.x;              // 0..63

    xs[t]       = vc[b * NUM_M + t];
    xs[t + 64]  = vc[b * NUM_M + t + 64];
    xs[t + 128] = vt[b * NUM_M + t];
    xs[t + 192] = vt[b * NUM_M + t + 64];
    __syncthreads();

    float h = b1[t];
    for (int i = 0; i < 2 * NUM_M; ++i)
        h += xs[i] * W1[i * HID + t];
    hs[t] = fmaxf(h, 0.0f);
    __syncthreads();

    float z = b2[t];
    for (int j = 0; j < HID; ++j)
        z += hs[j] * W2[j * NUM_T + t];
    alphas_out[b * NUM_T + t] = 1.0f / (1.0f + expf(-z));
}

// ---------------------------------------------------------------------------
// 128x128x128 f32 matmul, result kept in registers (wave w -> rows 16w..16w+15)
// via V_WMMA_F32_16X16X4_F32.  Fragment layouts per ISA 7.12.2 (32-bit):
//   A 16x4 : a.x = A[m][k + 2*half], a.y = A[m][k + 2*half + 1]
//   B 4x16 : b.x = B[k + 2*half][n], b.y = B[k + 2*half + 1][n]
//   C 16x16: acc[n][v] -> row = 16w + v + 8*half, col = 16n + m
// ---------------------------------------------------------------------------
__device__ __forceinline__ void mm128(const float* __restrict__ A,
                                      const float* __restrict__ B,
                                      v8f acc[8], int wave, int m, int half)
{
    #pragma unroll
    for (int n = 0; n < 8; ++n)
        acc[n] = (v8f){0.f,0.f,0.f,0.f,0.f,0.f,0.f,0.f};

    const int arow = (16 * wave + m) * LDPAD + 2 * half;
    #pragma unroll 4
    for (int kt = 0; kt < 32; ++kt) {
        const int kb = kt * 4;
        v2f a;
        a.x = A[arow + kb];
        a.y = A[arow + kb + 1];
        const float* B0 = B + (kb + 2 * half) * LDPAD + m;
        #pragma unroll
        for (int n = 0; n < 8; ++n) {
            v2f bb;
            bb.x = B0[n * 16];
            bb.y = B0[n * 16 + LDPAD];
            acc[n] = __builtin_amdgcn_wmma_f32_16x16x4_f32(
                false, a, false, bb, (short)0, acc[n], false, false);
        }
    }
}

// ---------------------------------------------------------------------------
// expm + matvec + store for one batch element whose pre-scaled generator
// X = A/16 already sits in LDS buffer Xsrc (LDPAD layout).
//   T = I + X/8;  7x Horner: T = I + (X*T)/k;  4x squaring: T = T*T
//   v_final = R @ v_current
// ---------------------------------------------------------------------------
__device__ __forceinline__ void expm_and_out(
    const float* __restrict__ Xsrc, float* __restrict__ Ts,
    float* __restrict__ vs, float* __restrict__ partial,
    const float* __restrict__ vcur, float* __restrict__ vfinal,
    int b, int tid, int wave, int m, int half)
{
    // ---- seed T = I + X/8 (order-8 Horner seed); thread owns row tid>>1
    {
        const int r  = tid >> 1;
        const int cb = (tid & 1) << 6;
        #pragma unroll 16
        for (int i = 0; i < 64; ++i) {
            const int c = cb + i;
            const float x = Xsrc[r * LDPAD + c];
            Ts[r * LDPAD + c] = x * 0.125f + ((r == c) ? 1.0f : 0.0f);
        }
    }
    __syncthreads();

    v8f accf[8];

    // ---- Horner: k = 7..1 : T = I + (X*T)/k   (result in registers, so T is
    //      updated in place after one barrier)
    for (int step = 0; step < 7; ++step) {
        mm128(Xsrc, Ts, accf, wave, m, half);
        __syncthreads();                       // all reads of Ts complete
        const float ik = 1.0f / (float)(7 - step);
        #pragma unroll
        for (int n = 0; n < 8; ++n)
            #pragma unroll
            for (int v = 0; v < 8; ++v) {
                const int row = 16 * wave + v + 8 * half;
                const int col = 16 * n + m;
                Ts[row * LDPAD + col] =
                    accf[n][v] * ik + ((row == col) ? 1.0f : 0.0f);
            }
        __syncthreads();
    }

    // ---- squarings: T = T*T, 4 times  -> R = expm(A)
    for (int s = 0; s < 4; ++s) {
        mm128(Ts, Ts, accf, wave, m, half);
        __syncthreads();
        #pragma unroll
        for (int n = 0; n < 8; ++n)
            #pragma unroll
            for (int v = 0; v < 8; ++v) {
                const int row = 16 * wave + v + 8 * half;
                const int col = 16 * n + m;
                Ts[row * LDPAD + col] = accf[n][v];
            }
        __syncthreads();
    }

    // ---- v_final = R @ v_current : 2 threads per row
    if (tid < NUM_M) vs[tid] = vcur[(size_t)b * NUM_M + tid];
    __syncthreads();
    {
        const int r  = tid >> 1;
        const int cb = (tid & 1) << 6;
        float s = 0.0f;
        #pragma unroll 8
        for (int i = 0; i < 64; ++i)
            s += Ts[r * LDPAD + cb + i] * vs[cb + i];
        partial[tid] = s;
    }
    __syncthreads();                           // Ts/vs reads done; partial ready
    if (tid < NUM_M)
        vfinal[(size_t)b * NUM_M + tid] = partial[2 * tid] + partial[2 * tid + 1];
    __syncthreads();                           // partial/Ts free for reuse
}

// ---------------------------------------------------------------------------
// Kernel 2: one WG handles FOUR batch elements (4g .. 4g+3).
//   Streaming reads each Omega float4 ONCE and feeds all four weighted sums
//   -> quarters the dominant L2 traffic (16 GB -> 4 GB total).
//   X0..X2 land in LDS (3 matrices + T = 272 KB of the 320 KB WGP LDS);
//   X3 rides in VGPRs only until expm #0 frees the Xs0 buffer.
// ---------------------------------------------------------------------------
__global__ __launch_bounds__(256) void expm_kernel(
    const float* __restrict__ vcur, const float* __restrict__ omegas,
    const float* __restrict__ alphas, float* __restrict__ vfinal)
{
    __shared__ float Xs0[NUM_M * LDPAD];
    __shared__ float Xs1[NUM_M * LDPAD];
    __shared__ float Xs2[NUM_M * LDPAD];
    __shared__ float Ts [NUM_M * LDPAD];
    __shared__ float al [GRP * NUM_T];
    __shared__ float vs [NUM_M];
    __shared__ float partial[256];

    const int g    = blockIdx.x;
    const int b0   = GRP * g;
    const int tid  = threadIdx.x;
    const int wave = tid >> 5;
    const int lane = tid & 31;
    const int m    = lane & 15;
    const int half = lane >> 4;

    // alphas for b0..b0+3 are 256 contiguous floats
    al[tid] = alphas[(size_t)b0 * NUM_T + tid];
    __syncthreads();

    // ---- joint weighted sum: each thread owns 64 contiguous matrix elements
    float4 a0[16], a1[16], a2[16], a3[16];
    #pragma unroll
    for (int j = 0; j < 16; ++j) {
        a0[j] = make_float4(0.f, 0.f, 0.f, 0.f);
        a1[j] = make_float4(0.f, 0.f, 0.f, 0.f);
        a2[j] = make_float4(0.f, 0.f, 0.f, 0.f);
        a3[j] = make_float4(0.f, 0.f, 0.f, 0.f);
    }

    const float4* om4 = reinterpret_cast<const float4*>(omegas);
    for (int t = 0; t < NUM_T; ++t) {
        const float s0 = al[t];
        const float s1 = al[NUM_T + t];
        const float s2 = al[2 * NUM_T + t];
        const float s3 = al[3 * NUM_T + t];
        const float4* p = om4 + (size_t)t * 4096 + (size_t)tid * 16;
        #pragma unroll
        for (int j = 0; j < 16; ++j) {
            float4 w = p[j];
            a0[j].x += s0 * w.x;  a0[j].y += s0 * w.y;
            a0[j].z += s0 * w.z;  a0[j].w += s0 * w.w;
            a1[j].x += s1 * w.x;  a1[j].y += s1 * w.y;
            a1[j].z += s1 * w.z;  a1[j].w += s1 * w.w;
            a2[j].x += s2 * w.x;  a2[j].y += s2 * w.y;
            a2[j].z += s2 * w.z;  a2[j].w += s2 * w.w;
            a3[j].x += s3 * w.x;  a3[j].y += s3 * w.y;
            a3[j].z += s3 * w.z;  a3[j].w += s3 * w.w;
        }
    }

    // ---- write X0..X2 = A/16 to LDS; X3 stays in registers for now
    {
        const int r  = tid >> 1;
        const int cb = (tid & 1) << 6;
        #pragma unroll
        for (int j = 0; j < 16; ++j) {
            const float v0[4] = {a0[j].x, a0[j].y, a0[j].z, a0[j].w};
            const float v1[4] = {a1[j].x, a1[j].y, a1[j].z, a1[j].w};
            const float v2[4] = {a2[j].x, a2[j].y, a2[j].z, a2[j].w};
            #pragma unroll
            for (int c2 = 0; c2 < 4; ++c2) {
                const int c = cb + j * 4 + c2;
                Xs0[r * LDPAD + c] = v0[c2] * 0.0625f;     // 2^-4 scaling
                Xs1[r * LDPAD + c] = v1[c2] * 0.0625f;
                Xs2[r * LDPAD + c] = v2[c2] * 0.0625f;
            }
        }
    }
    __syncthreads();

    expm_and_out(Xs0, Ts, vs, partial, vcur, vfinal, b0 + 0, tid, wave, m, half);

    // expm #0 no longer needs Xs0 -> park X3 there (visibility guaranteed by
    // the barriers inside the following expm calls before Xs0 is read again)
    {
        const int r  = tid >> 1;
        const int cb = (tid & 1) << 6;
        #pragma unroll
        for (int j = 0; j < 16; ++j) {
            const float v3[4] = {a3[j].x, a3[j].y, a3[j].z, a3[j].w};
            #pragma unroll
            for (int c2 = 0; c2 < 4; ++c2)
                Xs0[r * LDPAD + cb + j * 4 + c2] = v3[c2] * 0.0625f;
        }
    }

    expm_and_out(Xs1, Ts, vs, partial, vcur, vfinal, b0 + 1, tid, wave, m, half);
    expm_and_out(Xs2, Ts, vs, partial, vcur, vfinal, b0 + 2, tid, wave, m, half);
    expm_and_out(Xs0, Ts, vs, partial, vcur, vfinal, b0 + 3, tid, wave, m, half);
}

// ---------------------------------------------------------------------------
extern "C" void kernel_launch(void* const* d_in, const int* in_sizes, int n_in,
                              void* d_out, int out_size, void* d_ws, size_t ws_size,
                              hipStream_t stream)
{
    (void)in_sizes; (void)n_in; (void)out_size; (void)d_ws; (void)ws_size;
    const float* vc = (const float*)d_in[0];
    const float* vt = (const float*)d_in[1];
    const float* W1 = (const float*)d_in[2];
    const float* b1 = (const float*)d_in[3];
    const float* W2 = (const float*)d_in[4];
    const float* b2 = (const float*)d_in[5];
    const float* om = (const float*)d_in[6];

    float* out    = (float*)d_out;
    float* vfinal = out;                         // [B, 128]
    float* alph   = out + (size_t)B_SZ * NUM_M;  // [B, 64]  (2nd tuple output)

    alphas_kernel<<<dim3(B_SZ), dim3(64), 0, stream>>>(vc, vt, W1, b1, W2, b2, alph);
    expm_kernel<<<dim3(B_SZ / GRP), dim3(256), 0, stream>>>(vc, om, alph, vfinal);
}